// RGCNLayer_33122787786775
// MI455X (gfx1250) — compile-verified
//
#include <hip/hip_runtime.h>

// RGCN layer for MI455X (gfx1250, wave32).
// Pipeline:
//   1) counting-sort edges by relation (16 bins)
//   2) one wave per 16-edge tile: gather x[src] rows -> LDS,
//      V_WMMA_F32_16X16X4_F32 against W[rel] (16x128 @ 128x128)
//   3a) BIG ws:  store per-edge message rows, counting-sort edge ids by
//       target (50k bins, parallel scan), then one wave per target gathers
//       + register-reduces its messages, ReLU, single store.  (no f32 atomics)
//   3b) SMALL ws: atomic-scatter messages into out[tgt], then ReLU pass.

typedef __attribute__((ext_vector_type(2))) float v2f;
typedef __attribute__((ext_vector_type(8))) float v8f;

#define FEAT 128
#define TILE 16
#define WPB  4      // waves (tiles) per block in the WMMA kernel
#define APAD 132    // LDS row stride in floats: 4*row distinct banks, 8B aligned

// ---------------- generic helpers ----------------

__global__ void zero_f32(float* __restrict__ p, int n) {
  for (int i = blockIdx.x * blockDim.x + threadIdx.x; i < n; i += gridDim.x * blockDim.x)
    p[i] = 0.0f;
}

__global__ void zero_u32(unsigned* __restrict__ p, int n) {
  for (int i = blockIdx.x * blockDim.x + threadIdx.x; i < n; i += gridDim.x * blockDim.x)
    p[i] = 0u;
}

// ---------------- relation counting sort (16 bins) ----------------

__global__ void hist_rel(const int* __restrict__ etype, int E, unsigned* __restrict__ hist) {
  int e = blockIdx.x * blockDim.x + threadIdx.x;
  if (e < E) atomicAdd(&hist[etype[e] & 15], 1u);
}

__global__ void scan_rel(const unsigned* __restrict__ hist, unsigned* __restrict__ cursor) {
  if (threadIdx.x == 0) {
    unsigned acc = 0;
    for (int r = 0; r < 16; ++r) { cursor[r] = acc; acc += hist[r]; }
  }
}

__global__ void scatter_rel(const int* __restrict__ eidx, const int* __restrict__ etype,
                            int E, unsigned* __restrict__ cursor,
                            int* __restrict__ ssrc, int* __restrict__ stgt,
                            int* __restrict__ srel) {
  int e = blockIdx.x * blockDim.x + threadIdx.x;
  if (e < E) {
    int r = etype[e] & 15;
    unsigned p = atomicAdd(&cursor[r], 1u);
    ssrc[p] = eidx[e];        // edge_index[0][e]
    stgt[p] = eidx[E + e];    // edge_index[1][e]
    srel[p] = r;
  }
}

// ---------------- target counting sort (N bins, parallel scan) ----------------

__global__ void hist_tgt(const int* __restrict__ stgt, int E, unsigned* __restrict__ hist2) {
  int e = blockIdx.x * blockDim.x + threadIdx.x;
  if (e < E) atomicAdd(&hist2[stgt[e]], 1u);
}

__global__ void blocksum_kernel(const unsigned* __restrict__ hist2, int n,
                                unsigned* __restrict__ bsum) {
  __shared__ unsigned red[256];
  int i = blockIdx.x * 256 + threadIdx.x;
  red[threadIdx.x] = (i < n) ? hist2[i] : 0u;
  __syncthreads();
  for (int s = 128; s > 0; s >>= 1) {
    if (threadIdx.x < s) red[threadIdx.x] += red[threadIdx.x + s];
    __syncthreads();
  }
  if (threadIdx.x == 0) bsum[blockIdx.x] = red[0];
}

__global__ void scan_blocks(unsigned* __restrict__ bsum, int nb,
                            unsigned* __restrict__ off2, int n) {
  if (threadIdx.x == 0) {
    unsigned acc = 0;
    for (int b = 0; b < nb; ++b) { unsigned t = bsum[b]; bsum[b] = acc; acc += t; }
    off2[n] = acc;  // == E
  }
}

__global__ void scan_final(const unsigned* __restrict__ hist2,
                           const unsigned* __restrict__ bsum, int n,
                           unsigned* __restrict__ off2, unsigned* __restrict__ cursor2) {
  __shared__ unsigned tmp[256];
  int i = blockIdx.x * 256 + threadIdx.x;
  unsigned v = (i < n) ? hist2[i] : 0u;
  tmp[threadIdx.x] = v;
  __syncthreads();
  for (int d = 1; d < 256; d <<= 1) {             // Hillis-Steele inclusive scan
    unsigned t = (threadIdx.x >= (unsigned)d) ? tmp[threadIdx.x - d] : 0u;
    __syncthreads();
    tmp[threadIdx.x] += t;
    __syncthreads();
  }
  unsigned excl = tmp[threadIdx.x] - v + bsum[blockIdx.x];
  if (i < n) { off2[i] = excl; cursor2[i] = excl; }
}

__global__ void scatter_tgt(const int* __restrict__ stgt, int E,
                            unsigned* __restrict__ cursor2, int* __restrict__ perm) {
  int e = blockIdx.x * blockDim.x + threadIdx.x;
  if (e < E) {
    unsigned p = atomicAdd(&cursor2[stgt[e]], 1u);
    perm[p] = e;  // msg row id (relation-sorted order)
  }
}

// ---------------- main WMMA tile kernel ----------------

__global__ __launch_bounds__(WPB * 32) void rgcn_tile_kernel(
    const float* __restrict__ x, const float* __restrict__ w,
    const int* __restrict__ ssrc, const int* __restrict__ stgt,
    const int* __restrict__ srel, int E,
    float* __restrict__ msg,   // big path: per-edge message rows; null -> atomic mode
    float* __restrict__ out) {

  __shared__ float sA[WPB][TILE * APAD];  // gathered 16x128 A tiles (padded)
  __shared__ int   sT[WPB][TILE];         // target node per row
  __shared__ int   sR[WPB][TILE];         // relation per row (-1 = invalid)

  const int lane = threadIdx.x & 31;
  const int wv   = threadIdx.x >> 5;
  const int tile = blockIdx.x * WPB + wv;
  const int e0   = tile * TILE;
  const bool tvalid = (e0 < E);

  // ---- phase 1: gather edge meta + 16 source rows into LDS ----
  int srcv = 0, relv = -1;
  if (tvalid && lane < TILE) {
    int e = e0 + lane;
    int tg = -1;
    if (e < E) { srcv = ssrc[e]; tg = stgt[e]; relv = srel[e]; }
    sT[wv][lane] = tg;
    sR[wv][lane] = relv;
  }
  if (tvalid) {
    #pragma unroll 4
    for (int m = 0; m < TILE; ++m) {
      int sm = __shfl(srcv, m);
      int rm = __shfl(relv, m);
      v2f lo = {0.f, 0.f}, hi = {0.f, 0.f};
      if (rm >= 0) {
        const float* xr = x + (size_t)sm * FEAT + lane * 4;
        lo = *(const v2f*)(xr);
        hi = *(const v2f*)(xr + 2);
      }
      float* dst = &sA[wv][m * APAD + lane * 4];   // byte addr: m*528 + lane*16 (8B ok)
      *(v2f*)(dst)     = lo;
      *(v2f*)(dst + 2) = hi;
    }
  }
  __syncthreads();

  if (!tvalid) return;

  // ---- phase 2: per relation-uniform segment, WMMA + emit ----
  const int row   = lane & 15;        // A row (M) this lane feeds
  const int col   = lane & 15;        // D/B column (N) this lane holds
  const int kh    = (lane >> 4) << 1; // K sub-offset (0 or 2)
  const int mhalf = (lane >> 4) << 3; // D row half (0 or 8)

  int s = 0;
  while (s < TILE) {
    int r = sR[wv][s];
    int t = s + 1;
    while (t < TILE && sR[wv][t] == r) ++t;
    if (r >= 0) {
      const float* Wr = w + (size_t)r * FEAT * FEAT;
      const float msk = (row >= s && row < t) ? 1.0f : 0.0f;

      v8f acc[8];
      #pragma unroll
      for (int n = 0; n < 8; ++n) acc[n] = (v8f)(0.0f);

      for (int k0 = 0; k0 < FEAT; k0 += 4) {
        v2f a = *(const v2f*)&sA[wv][row * APAD + k0 + kh];
        a = a * msk;  // zero rows outside this segment (VALU select: EXEC stays full)
        const float* bp = Wr + (size_t)(k0 + kh) * FEAT + col;
        #pragma unroll
        for (int n = 0; n < 8; ++n) {
          v2f b = { bp[n * 16], bp[n * 16 + FEAT] };
          acc[n] = __builtin_amdgcn_wmma_f32_16x16x4_f32(
              false, a, false, b, (short)0, acc[n], false, false);
        }
      }

      // emit 16x128 message tile: lane holds D(M = g+mhalf, N = col)
      if (msg) {                       // store mode (no atomics)
        #pragma unroll
        for (int g = 0; g < 8; ++g) {
          int M = g + mhalf;
          if (M >= s && M < t) {
            float* op = msg + (size_t)(e0 + M) * FEAT + col;
            #pragma unroll
            for (int n = 0; n < 8; ++n) op[n * 16] = acc[n][g];
          }
        }
      } else {                         // fallback: atomic scatter
        #pragma unroll
        for (int g = 0; g < 8; ++g) {
          int M = g + mhalf;
          if (M >= s && M < t) {
            int tg = sT[wv][M];
            float* op = out + (size_t)tg * FEAT + col;
            #pragma unroll
            for (int n = 0; n < 8; ++n)
              unsafeAtomicAdd(op + n * 16, acc[n][g]);
          }
        }
      }
    }
    s = t;
  }
}

// ---------------- gather-reduce (big path) ----------------

__global__ __launch_bounds__(256) void reduce_kernel(
    const float* __restrict__ msg, const int* __restrict__ perm,
    const unsigned* __restrict__ off2, int n, float* __restrict__ out) {
  int wid  = (blockIdx.x * 256 + threadIdx.x) >> 5;  // one wave per target node
  int lane = threadIdx.x & 31;
  if (wid >= n) return;
  unsigned b = off2[wid], e = off2[wid + 1];
  float a0 = 0.f, a1 = 0.f, a2 = 0.f, a3 = 0.f;
  for (unsigned i = b; i < e; ++i) {
    const float* mp = msg + (size_t)perm[i] * FEAT + lane * 4;  // 512B/wave, coalesced
    a0 += mp[0]; a1 += mp[1]; a2 += mp[2]; a3 += mp[3];
  }
  float* op = out + (size_t)wid * FEAT + lane * 4;
  op[0] = fmaxf(a0, 0.f); op[1] = fmaxf(a1, 0.f);
  op[2] = fmaxf(a2, 0.f); op[3] = fmaxf(a3, 0.f);
}

// ---------------- epilogue (fallback path) ----------------

__global__ void relu_kernel(float* __restrict__ out, int n) {
  for (int i = blockIdx.x * blockDim.x + threadIdx.x; i < n; i += gridDim.x * blockDim.x)
    out[i] = fmaxf(out[i], 0.0f);
}

// ---------------- launcher ----------------

extern "C" void kernel_launch(void* const* d_in, const int* in_sizes, int n_in,
                              void* d_out, int out_size, void* d_ws, size_t ws_size,
                              hipStream_t stream) {
  const float* x     = (const float*)d_in[0];   // (N, 128) f32
  const float* w     = (const float*)d_in[1];   // (16, 128, 128) f32
  const int*   eidx  = (const int*)d_in[2];     // (2, E) int
  const int*   etype = (const int*)d_in[3];     // (E,) int
  float* out = (float*)d_out;                   // (N, 128) f32

  const int N = in_sizes[0] / FEAT;
  const int E = in_sizes[3];
  const int nb = (N + 255) / 256;               // scan blocks over N bins

  // workspace carve-out
  char* base = (char*)d_ws;
  size_t off = 0;
  auto alloc = [&](size_t bytes, size_t align) -> size_t {
    off = (off + align - 1) / align * align;
    size_t r = off; off += bytes; return r;
  };
  unsigned* hist    = (unsigned*)(base + alloc(16 * 4, 4));
  unsigned* cursor  = (unsigned*)(base + alloc(16 * 4, 4));
  int*      ssrc    = (int*)     (base + alloc((size_t)E * 4, 4));
  int*      stgt    = (int*)     (base + alloc((size_t)E * 4, 4));
  int*      srel    = (int*)     (base + alloc((size_t)E * 4, 4));
  size_t small_need = off;                       // fallback needs only this much
  unsigned* hist2   = (unsigned*)(base + alloc((size_t)N * 4, 4));
  unsigned* cursor2 = (unsigned*)(base + alloc((size_t)N * 4, 4));
  unsigned* off2    = (unsigned*)(base + alloc(((size_t)N + 1) * 4, 4));
  unsigned* bsum    = (unsigned*)(base + alloc((size_t)nb * 4, 4));
  int*      perm    = (int*)     (base + alloc((size_t)E * 4, 4));
  float*    msg     = (float*)   (base + alloc((size_t)E * FEAT * 4, 256));
  const bool big = (ws_size >= off) && (ws_size >= small_need);

  // --- relation counting sort (both paths) ---
  zero_u32<<<1, 64, 0, stream>>>(hist, 16);
  hist_rel<<<(E + 255) / 256, 256, 0, stream>>>(etype, E, hist);
  scan_rel<<<1, 32, 0, stream>>>(hist, cursor);
  scatter_rel<<<(E + 255) / 256, 256, 0, stream>>>(eidx, etype, E, cursor, ssrc, stgt, srel);

  const int nTiles = (E + TILE - 1) / TILE;

  if (big) {
    // --- WMMA -> message store, target sort, gather-reduce ---
    zero_u32<<<(N + 255) / 256, 256, 0, stream>>>(hist2, N);
    rgcn_tile_kernel<<<(nTiles + WPB - 1) / WPB, WPB * 32, 0, stream>>>(
        x, w, ssrc, stgt, srel, E, msg, out);
    hist_tgt<<<(E + 255) / 256, 256, 0, stream>>>(stgt, E, hist2);
    blocksum_kernel<<<nb, 256, 0, stream>>>(hist2, N, bsum);
    scan_blocks<<<1, 32, 0, stream>>>(bsum, nb, off2, N);
    scan_final<<<nb, 256, 0, stream>>>(hist2, bsum, N, off2, cursor2);
    scatter_tgt<<<(E + 255) / 256, 256, 0, stream>>>(stgt, E, cursor2, perm);
    reduce_kernel<<<((size_t)N * 32 + 255) / 256, 256, 0, stream>>>(msg, perm, off2, N, out);
  } else {
    // --- WMMA -> atomic scatter, ReLU ---
    zero_f32<<<2048, 256, 0, stream>>>(out, out_size);
    rgcn_tile_kernel<<<(nTiles + WPB - 1) / WPB, WPB * 32, 0, stream>>>(
        x, w, ssrc, stgt, srel, E, (float*)nullptr, out);
    relu_kernel<<<2048, 256, 0, stream>>>(out, out_size);
  }
}